// SceneGNN_57380763075218
// MI455X (gfx1250) — compile-verified
//
#include <hip/hip_runtime.h>
#include <math.h>

#define Nn   50000
#define Ee   800000
#define ETOT (Nn + Ee)
#define Bb   64
#define HIDD 128
#define NH   4
#define CD   32
#define LAY  3

#define LDS_STRIDE 132  // 128 + 4 pad -> conflict-free A-fragment reads (bank step 4)

typedef __attribute__((ext_vector_type(2))) float v2f;
typedef __attribute__((ext_vector_type(8))) float v8f;

// ---- ordered-uint encoding so unsigned atomicMax == float max ----
__device__ __forceinline__ unsigned ordf(float f) {
  unsigned u = __float_as_uint(f);
  return (u & 0x80000000u) ? ~u : (u | 0x80000000u);
}
__device__ __forceinline__ float unordf(unsigned u) {
  unsigned v = (u & 0x80000000u) ? (u & 0x7FFFFFFFu) : ~u;
  return __uint_as_float(v);
}

// ---------------- generic zero ----------------
__global__ void k_zero(float* p, long n) {
  long i = (long)blockIdx.x * blockDim.x + threadIdx.x;
  long st = (long)gridDim.x * blockDim.x;
  for (; i < n; i += st) p[i] = 0.0f;
}

// ---------------- edge_attr column sums ----------------
__global__ void k_ea_sum(const float* __restrict__ ea, float* __restrict__ sum) {
  __shared__ float s[16];
  int t = threadIdx.x;
  if (t < 16) s[t] = 0.0f;
  __syncthreads();
  float loc[16];
#pragma unroll
  for (int c = 0; c < 16; ++c) loc[c] = 0.0f;
  for (long e = (long)blockIdx.x * blockDim.x + t; e < Ee;
       e += (long)gridDim.x * blockDim.x) {
    const float* row = ea + e * 16;
#pragma unroll
    for (int c = 0; c < 16; ++c) loc[c] += row[c];
  }
#pragma unroll
  for (int c = 0; c < 16; ++c) atomicAdd(&s[c], loc[c]);
  __syncthreads();
  if (t < 16) atomicAdd(&sum[t], s[t]);
}

// ---------------- fold We and a_e: weAtt[l][k][h], loopAtt[l][h] ----------------
__global__ void k_prep(const float* __restrict__ eaSum, const float* __restrict__ linEW,
                       const float* __restrict__ attE, float* __restrict__ eaMean,
                       float* __restrict__ weAtt, float* __restrict__ loopAtt) {
  int t = threadIdx.x;
  if (t < 16) eaMean[t] = eaSum[t] * (1.0f / (float)Ee);
  __syncthreads();
  if (t < LAY * 16 * NH) {
    int l = t / (16 * NH);
    int r = t % (16 * NH);
    int k = r / NH, h = r % NH;
    const float* we = linEW + ((long)l * 16 + k) * HIDD + h * CD;
    const float* ae = attE + (l * NH + h) * CD;
    float s = 0.0f;
#pragma unroll
    for (int c = 0; c < CD; ++c) s += we[c] * ae[c];
    weAtt[t] = s;
  }
  __syncthreads();
  if (t < LAY * NH) {
    int l = t / NH, h = t % NH;
    float s = 0.0f;
#pragma unroll
    for (int k = 0; k < 16; ++k) s += eaMean[k] * weAtt[(l * 16 + k) * NH + h];
    loopAtt[t] = s;
  }
}

// ---------------- h0 = relu(x @ projW + b) ----------------
__global__ void k_proj(const float* __restrict__ x, const float* __restrict__ W,
                       const float* __restrict__ b, float* __restrict__ h) {
  long i = (long)blockIdx.x * blockDim.x + threadIdx.x;
  if (i >= (long)Nn * HIDD) return;
  int n = (int)(i >> 7), c = (int)(i & 127);
  const float* xr = x + (long)n * 8;
  float s = b[c];
#pragma unroll
  for (int k = 0; k < 8; ++k) s += xr[k] * W[k * HIDD + c];
  h[i] = fmaxf(s, 0.0f);
}

// ---------------- WMMA f32 GEMM: Y[n,128] = A[n,128] @ W[128,128] ----------------
// One block (8 waves) owns a 16-row stripe of A: the stripe is staged to LDS ONCE
// via GLOBAL_LOAD_ASYNC_TO_LDS_B128 (ASYNCcnt), then each wave computes one 16x16
// column tile with V_WMMA_F32_16X16X4_F32, A-fragments read from LDS (ds_load).
__global__ void __launch_bounds__(256) k_gemm128(const float* __restrict__ A,
                                                 const float* __restrict__ W,
                                                 float* __restrict__ Y, int nRows) {
  __shared__ float sA[16 * LDS_STRIDE];  // padded 16x128 stripe
  int row0 = blockIdx.x * 16;
  if (row0 >= nRows) return;

  // LDS byte offset of sA: generic LDS pointers carry the ds offset in low 32 bits.
  unsigned ldsBase = (unsigned)(unsigned long long)(uintptr_t)&sA[0];

  // Stage A[row0 .. row0+16) x 128 into LDS: 512 chunks of 16B, 2 per thread.
  for (int i = threadIdx.x; i < 512; i += 256) {
    int row = i >> 5;           // 32 x 16B chunks per 128-float row
    int colf = (i & 31) * 4;    // starting float column of this chunk
    unsigned ldsOff = ldsBase + (unsigned)(row * LDS_STRIDE + colf) * 4u;
    const float* gp = A + ((long)(row0 + row)) * HIDD + colf;
    asm volatile("global_load_async_to_lds_b128 %0, %1, off"
                 :: "v"(ldsOff), "v"(gp) : "memory");
  }
  asm volatile("s_wait_asynccnt 0" ::: "memory");
  __syncthreads();

  int wave = threadIdx.x >> 5;  // 0..7 -> column tile
  int lane = threadIdx.x & 31;
  int col0 = wave * 16;
  int lm = lane & 15;
  bool hi = lane >= 16;  // upper half-wave carries K+2/K+3 (A) and M+8 (C/D)
  int col = col0 + lm;

  v8f c = {};
  for (int k = 0; k < HIDD; k += 4) {
    int kk = k + (hi ? 2 : 0);
    v2f a, bfrag;
    a.x = sA[lm * LDS_STRIDE + kk];
    a.y = sA[lm * LDS_STRIDE + kk + 1];
    bfrag.x = W[kk * HIDD + col];
    bfrag.y = W[(kk + 1) * HIDD + col];
    // 8 args: (neg_a, A, neg_b, B, c_mod, C, reuse_a, reuse_b)
    c = __builtin_amdgcn_wmma_f32_16x16x4_f32(false, a, false, bfrag, (short)0, c,
                                              false, false);
  }
  int m0 = row0 + (hi ? 8 : 0);
#pragma unroll
  for (int v = 0; v < 8; ++v) Y[(long)(m0 + v) * HIDD + col] = c[v];
}

// ---------------- per-node attention dots ----------------
__global__ void k_attdot(const float* __restrict__ xh, const float* __restrict__ as,
                         const float* __restrict__ ad, float* __restrict__ aS,
                         float* __restrict__ aD) {
  int i = blockIdx.x * blockDim.x + threadIdx.x;  // n*4 + h
  if (i >= Nn * NH) return;
  int n = i >> 2, h = i & 3;
  const float* r = xh + (long)n * HIDD + h * CD;
  float s = 0.0f, d = 0.0f;
#pragma unroll
  for (int c = 0; c < CD; ++c) {
    float v = r[c];
    s += v * as[h * CD + c];
    d += v * ad[h * CD + c];
  }
  aS[i] = s;
  aD[i] = d;
}

// ---------------- edge pass 1: logits + segment max ----------------
__global__ void k_e1(const int* __restrict__ ei, const float* __restrict__ ea,
                     const float* __restrict__ aS, const float* __restrict__ aD,
                     const float* __restrict__ weAtt, const float* __restrict__ loopAtt,
                     float* __restrict__ alpha, unsigned* __restrict__ amax) {
  long e = (long)blockIdx.x * blockDim.x + threadIdx.x;
  if (e >= ETOT) return;
  int s, d;
  float ec[NH];
  if (e < Ee) {
    s = ei[e];
    d = ei[Ee + e];
    const float* row = ea + e * 16;
    float t0 = 0, t1 = 0, t2 = 0, t3 = 0;
#pragma unroll
    for (int k = 0; k < 16; ++k) {
      float v = row[k];
      t0 += v * weAtt[k * NH + 0];
      t1 += v * weAtt[k * NH + 1];
      t2 += v * weAtt[k * NH + 2];
      t3 += v * weAtt[k * NH + 3];
    }
    ec[0] = t0; ec[1] = t1; ec[2] = t2; ec[3] = t3;
  } else {
    s = d = (int)(e - Ee);
#pragma unroll
    for (int h = 0; h < NH; ++h) ec[h] = loopAtt[h];
  }
#pragma unroll
  for (int h = 0; h < NH; ++h) {
    float v = aS[s * NH + h] + aD[d * NH + h] + ec[h];
    v = (v > 0.0f) ? v : 0.2f * v;  // leaky_relu 0.2
    alpha[e * NH + h] = v;
    atomicMax(&amax[(long)d * NH + h], ordf(v));
  }
}

// ---------------- edge pass 2: exp + segment sum ----------------
__global__ void k_e2(const int* __restrict__ ei, float* __restrict__ alpha,
                     const unsigned* __restrict__ amax, float* __restrict__ denom) {
  long e = (long)blockIdx.x * blockDim.x + threadIdx.x;
  if (e >= ETOT) return;
  int d = (e < Ee) ? ei[Ee + e] : (int)(e - Ee);
#pragma unroll
  for (int h = 0; h < NH; ++h) {
    float a = expf(alpha[e * NH + h] - unordf(amax[(long)d * NH + h]));
    alpha[e * NH + h] = a;
    atomicAdd(&denom[(long)d * NH + h], a);
  }
}

// ---------------- edge pass 3: message scatter (wave per edge, 4 ch/lane) ----------------
__global__ void __launch_bounds__(256) k_e3(const int* __restrict__ ei,
                                            const float* __restrict__ xh,
                                            const float* __restrict__ alpha,
                                            const float* __restrict__ denom,
                                            float* __restrict__ acc) {
  long t = (long)blockIdx.x * blockDim.x + threadIdx.x;
  long e = t >> 5;
  if (e >= ETOT) return;
  int lane = (int)(t & 31);
  int s, d;
  if (e < Ee) { s = ei[e]; d = ei[Ee + e]; }
  else        { s = d = (int)(e - Ee); }
  __builtin_prefetch(xh + (long)s * HIDD + lane * 4, 0, 1);  // global_prefetch_b8
  int h = lane >> 3;  // channels [lane*4, lane*4+4) all belong to head lane/8
  float coef = alpha[e * NH + h] / (denom[(long)d * NH + h] + 1e-16f);
  const float4 v = *(const float4*)(xh + (long)s * HIDD + lane * 4);
  float* o = acc + (long)d * HIDD + lane * 4;
  atomicAdd(o + 0, v.x * coef);
  atomicAdd(o + 1, v.y * coef);
  atomicAdd(o + 2, v.z * coef);
  atomicAdd(o + 3, v.w * coef);
}

// ---------------- fused bias + residual + layernorm + relu (wave per node) ----------------
__global__ void __launch_bounds__(256) k_ln(const float* __restrict__ acc,
                                            const float* __restrict__ res,
                                            const float* __restrict__ bias,
                                            const float* __restrict__ g,
                                            const float* __restrict__ b,
                                            float* __restrict__ dst) {
  long t = (long)blockIdx.x * blockDim.x + threadIdx.x;
  long n = t >> 5;
  if (n >= Nn) return;
  int lane = (int)(t & 31);
  float v[4];
  float sum = 0.0f;
#pragma unroll
  for (int i = 0; i < 4; ++i) {
    int c = lane * 4 + i;
    v[i] = acc[n * HIDD + c] + bias[c] + res[n * HIDD + c];
    sum += v[i];
  }
#pragma unroll
  for (int m = 16; m > 0; m >>= 1) sum += __shfl_xor(sum, m, 32);
  float mu = sum * (1.0f / HIDD);
  float vs = 0.0f;
#pragma unroll
  for (int i = 0; i < 4; ++i) {
    float dd = v[i] - mu;
    vs += dd * dd;
  }
#pragma unroll
  for (int m = 16; m > 0; m >>= 1) vs += __shfl_xor(vs, m, 32);
  float inv = rsqrtf(vs * (1.0f / HIDD) + 1e-5f);
#pragma unroll
  for (int i = 0; i < 4; ++i) {
    int c = lane * 4 + i;
    float y = (v[i] - mu) * inv * g[c] + b[c];
    dst[n * HIDD + c] = fmaxf(y, 0.0f);
  }
}

// ---------------- pooling ----------------
__global__ void k_count(const int* __restrict__ batch, float* __restrict__ counts) {
  int n = blockIdx.x * blockDim.x + threadIdx.x;
  if (n >= Nn) return;
  atomicAdd(&counts[batch[n]], 1.0f);
}
__global__ void k_gsum(const int* __restrict__ batch, const float* __restrict__ h,
                       float* __restrict__ gAcc) {
  long i = (long)blockIdx.x * blockDim.x + threadIdx.x;
  if (i >= (long)Nn * HIDD) return;
  int n = (int)(i >> 7), c = (int)(i & 127);
  atomicAdd(&gAcc[(long)batch[n] * HIDD + c], h[i]);
}
__global__ void k_gfin(const float* __restrict__ gAcc, const float* __restrict__ counts,
                       float* __restrict__ outGraph) {
  int i = blockIdx.x * blockDim.x + threadIdx.x;
  if (i >= Bb * HIDD) return;
  outGraph[i] = gAcc[i] / fmaxf(counts[i >> 7], 1.0f);
}

// ---------------- MLP heads (one block per graph) ----------------
__global__ void __launch_bounds__(64) k_heads(const float* __restrict__ graph,
                                              const float* __restrict__ scW1,
                                              const float* __restrict__ scb1,
                                              const float* __restrict__ scW2,
                                              const float* __restrict__ scb2,
                                              const float* __restrict__ nvW1,
                                              const float* __restrict__ nvb1,
                                              const float* __restrict__ nvW2,
                                              const float* __restrict__ nvb2,
                                              float* __restrict__ scene,
                                              float* __restrict__ nav) {
  __shared__ float g[HIDD];
  __shared__ float hs[64], hn[64];
  int b = blockIdx.x, t = threadIdx.x;
  g[t] = graph[b * HIDD + t];
  g[t + 64] = graph[b * HIDD + t + 64];
  __syncthreads();
  float s = scb1[t], nv = nvb1[t];
  for (int c = 0; c < HIDD; ++c) {
    float gv = g[c];
    s += gv * scW1[c * 64 + t];
    nv += gv * nvW1[c * 64 + t];
  }
  hs[t] = fmaxf(s, 0.0f);
  hn[t] = fmaxf(nv, 0.0f);
  __syncthreads();
  if (t < 6) {
    float o = scb2[t];
    for (int j = 0; j < 64; ++j) o += hs[j] * scW2[j * 6 + t];
    scene[b * 6 + t] = o;
  }
  if (t < 4) {
    float o = nvb2[t];
    for (int j = 0; j < 64; ++j) o += hn[j] * nvW2[j * 4 + t];
    nav[b * 4 + t] = 1.0f / (1.0f + expf(-o));
  }
}

extern "C" void kernel_launch(void* const* d_in, const int* in_sizes, int n_in,
                              void* d_out, int out_size, void* d_ws, size_t ws_size,
                              hipStream_t stream) {
  const float* x     = (const float*)d_in[0];
  const int*   eidx  = (const int*)d_in[1];   // [2,E]
  const float* eattr = (const float*)d_in[2]; // [E,16]
  const int*   batch = (const int*)d_in[3];
  const float* projW = (const float*)d_in[4];
  const float* projB = (const float*)d_in[5];
  const float* linW  = (const float*)d_in[6];  // [3,128,128]
  const float* linEW = (const float*)d_in[7];  // [3,16,128]
  const float* attS  = (const float*)d_in[8];  // [3,4,32]
  const float* attD  = (const float*)d_in[9];
  const float* attE  = (const float*)d_in[10];
  const float* convB = (const float*)d_in[11]; // [3,128]
  const float* lnG   = (const float*)d_in[12];
  const float* lnB   = (const float*)d_in[13];
  const float* scW1  = (const float*)d_in[14];
  const float* scb1  = (const float*)d_in[15];
  const float* scW2  = (const float*)d_in[16];
  const float* scb2  = (const float*)d_in[17];
  const float* nvW1  = (const float*)d_in[18];
  const float* nvb1  = (const float*)d_in[19];
  const float* nvW2  = (const float*)d_in[20];
  const float* nvb2  = (const float*)d_in[21];

  float* outH     = (float*)d_out;
  float* outScene = outH + (long)Nn * HIDD;
  float* outNav   = outScene + Bb * 6;
  float* outGraph = outNav + Bb * 4;

  float* p = (float*)d_ws;
  float* hA      = p; p += (long)Nn * HIDD;
  float* xh      = p; p += (long)Nn * HIDD;
  float* acc     = p; p += (long)Nn * HIDD;   // acc, amax, denom contiguous -> one zero
  float* amaxF   = p; p += (long)Nn * NH;
  float* denom   = p; p += (long)Nn * NH;
  float* aS      = p; p += (long)Nn * NH;
  float* aD      = p; p += (long)Nn * NH;
  float* alpha   = p; p += (long)ETOT * NH;
  float* eaSum   = p; p += 16;
  float* eaMean  = p; p += 16;
  float* weAtt   = p; p += LAY * 16 * NH;
  float* loopAtt = p; p += LAY * NH;
  float* counts  = p; p += Bb;                // counts, gAcc contiguous -> one zero
  float* gAcc    = p; p += (long)Bb * HIDD;

  const int T = 256;
  const long nh = (long)Nn * HIDD;

  // edge-attr mean + folded attention weights
  k_zero<<<dim3(1), dim3(T), 0, stream>>>(eaSum, (long)16);
  k_ea_sum<<<dim3(512), dim3(T), 0, stream>>>(eattr, eaSum);
  k_prep<<<dim3(1), dim3(T), 0, stream>>>(eaSum, linEW, attE, eaMean, weAtt, loopAtt);

  // projection
  k_proj<<<dim3((unsigned)((nh + T - 1) / T)), dim3(T), 0, stream>>>(x, projW, projB, hA);

  for (int l = 0; l < LAY; ++l) {
    k_zero<<<dim3(4096), dim3(T), 0, stream>>>(acc, (long)Nn * HIDD + 2L * Nn * NH);

    int gemmBlocks = Nn / 16;  // one block per 16-row stripe (8 waves = 8 col tiles)
    k_gemm128<<<dim3(gemmBlocks), dim3(T), 0, stream>>>(hA, linW + (long)l * HIDD * HIDD,
                                                        xh, Nn);

    k_attdot<<<dim3((Nn * NH + T - 1) / T), dim3(T), 0, stream>>>(
        xh, attS + l * NH * CD, attD + l * NH * CD, aS, aD);

    k_e1<<<dim3((ETOT + T - 1) / T), dim3(T), 0, stream>>>(
        eidx, eattr, aS, aD, weAtt + l * 16 * NH, loopAtt + l * NH, alpha,
        (unsigned*)amaxF);
    k_e2<<<dim3((ETOT + T - 1) / T), dim3(T), 0, stream>>>(
        eidx, alpha, (const unsigned*)amaxF, denom);

    long e3t = (long)ETOT * 32;
    k_e3<<<dim3((unsigned)((e3t + T - 1) / T)), dim3(T), 0, stream>>>(eidx, xh, alpha,
                                                                      denom, acc);

    float* dst = (l == LAY - 1) ? outH : hA;
    long lnt = (long)Nn * 32;
    k_ln<<<dim3((unsigned)((lnt + T - 1) / T)), dim3(T), 0, stream>>>(
        acc, hA, convB + l * HIDD, lnG + l * HIDD, lnB + l * HIDD, dst);
  }

  // mean pooling + heads
  k_zero<<<dim3(64), dim3(T), 0, stream>>>(counts, (long)(Bb + Bb * HIDD));
  k_count<<<dim3((Nn + T - 1) / T), dim3(T), 0, stream>>>(batch, counts);
  k_gsum<<<dim3((unsigned)((nh + T - 1) / T)), dim3(T), 0, stream>>>(batch, outH, gAcc);
  k_gfin<<<dim3((Bb * HIDD + T - 1) / T), dim3(T), 0, stream>>>(gAcc, counts, outGraph);
  k_heads<<<dim3(Bb), dim3(64), 0, stream>>>(outGraph, scW1, scb1, scW2, scb2, nvW1,
                                             nvb1, nvW2, nvb2, outScene, outNav);
}